// InvariantMessagePassingTP_13649406066758
// MI455X (gfx1250) — compile-verified
//
#include <hip/hip_runtime.h>

// ---------------------------------------------------------------------------
// InvariantMessagePassingTP for MI455X (gfx1250, wave32)
//
// out[n,lm,c] = sum_{e: recv[e]==n} edge_attrs[e,lm] * tp_w[e,L(lm),c] * feats[n,c]
//
// Reformulated as a segmented GEMM so the inner loop is a single
// V_WMMA_F32_16X16X4_F32 per edge per 16-channel tile:
//   A[e]  (16x4):  A[lm][k] = edge_attrs[e,lm] * (k == L_OF_LM[lm])
//   B[e]  (4x16):  B[k][n]  = tp_w[e,k,ctile*16+n] * feats[recv[e], ctile*16+n]
//   C    (16x16):  running segment sum (receivers are sorted -> register-resident)
// Memory-bound problem (~550 MB @ 23.3 TB/s ~ 24 us); WMMA minimizes the
// per-byte instruction overhead vs a VALU formulation.
// ---------------------------------------------------------------------------

typedef float v2f __attribute__((ext_vector_type(2)));
typedef float v8f __attribute__((ext_vector_type(8)));

#define EPB   128   // edges per block (avg ~8 receiver runs per block)
#define NLM   16
#define NCH   64
#define WROW  (4 * NCH)   // floats per edge in tp_weights
#define PF_EDGES 8        // prefetch distance (edges) on the tp_weights stream

__global__ void __launch_bounds__(256)
imptp_zero(float* __restrict__ out, long long n) {
  long long i = (long long)blockIdx.x * blockDim.x + threadIdx.x;
  long long n4 = n >> 2;
  if (i < n4) ((float4*)out)[i] = make_float4(0.f, 0.f, 0.f, 0.f);
  if (i == 0) {
    for (long long t = (n4 << 2); t < n; ++t) out[t] = 0.f;
  }
}

__device__ __forceinline__ void flush_tile(float* __restrict__ out, const v8f acc,
                                           int node, int ctile, int lane,
                                           bool use_atomic) {
  // C/D layout (ISA 7.12.2): VGPR r holds M = r + 8*(lane>=16), N = lane&15
  const int n     = lane & 15;
  const int mbase = (lane >> 4) << 3;       // 0 or 8
  float* p = out + (size_t)node * (NLM * NCH) + (size_t)mbase * NCH + ctile * 16 + n;
  if (use_atomic) {
#pragma unroll
    for (int r = 0; r < 8; ++r) atomicAdd(p + r * NCH, acc[r]);
  } else {
#pragma unroll
    for (int r = 0; r < 8; ++r) p[r * NCH] = acc[r];
  }
}

__global__ void __launch_bounds__(128)
imptp_main(const float* __restrict__ node_feats,
           const float* __restrict__ edge_attrs,
           const float* __restrict__ tp_w,
           const int*   __restrict__ recv,
           float*       __restrict__ out,
           int nedges) {
  const int lane  = threadIdx.x & 31;
  const int ctile = threadIdx.x >> 5;                 // 0..3 -> channels [16*ctile, 16*ctile+16)
  const int e0 = blockIdx.x * EPB;
  if (e0 >= nedges) return;                           // uniform per block; EXEC stays all-1s
  const int e1 = min(e0 + EPB, nedges);

  // Per-lane constants.
  // A-matrix 16x4 f32 layout: lanes 0-15 hold K={0,1} (vgpr0,vgpr1) for M=lane;
  // lanes 16-31 hold K={2,3} for M=lane-16.  B 4x16 mirrors the K split with N on lanes.
  const int  m    = lane & 15;
  const int  Lm   = (m > 0) + (m > 3) + (m > 8);      // L_OF_LM[m]
  const int  khi  = (lane >> 4) << 1;                 // 0 (lanes 0-15) or 2 (lanes 16-31)
  const bool sel0 = (Lm == khi);
  const bool sel1 = (Lm == khi + 1);
  const int  c    = ctile * 16 + m;                   // this lane's channel (for B / feats)

  // Receiver runs: interior nodes are fully owned by this block (sorted, contiguous)
  // -> plain stores; first/last node may span blocks -> float atomics onto zeroed out.
  const int r_first = recv[e0];
  const int r_last  = recv[e1 - 1];

  v8f acc;
#pragma unroll
  for (int i = 0; i < 8; ++i) acc[i] = 0.f;
  int cur = r_first;

  for (int e = e0; e < e1; ++e) {
    const int r = recv[e];                            // uniform -> scalar load/branch
    if (r != cur) {
      flush_tile(out, acc, cur, ctile, lane, (cur == r_first) || (cur == r_last));
#pragma unroll
      for (int i = 0; i < 8; ++i) acc[i] = 0.f;
      cur = r;
    }

    // A: sparse embedding of edge_attrs row into the (lm, L) matrix
    const float ea = edge_attrs[(size_t)e * NLM + m];
    v2f A;
    A.x = sel0 ? ea : 0.f;
    A.y = sel1 ? ea : 0.f;

    // B: tp_w row k scaled by the gathered node feature (L2-resident: sorted recv)
    const float f  = node_feats[(size_t)r * NCH + c];
    const float* wp = tp_w + (size_t)e * WROW + (size_t)khi * NCH + c;
    v2f B;
    B.x = wp[0]   * f;
    B.y = wp[NCH] * f;

    // Prefetch the dominant stream (tp_weights: ~410 MB) -> global_prefetch_b8
    __builtin_prefetch(wp + (size_t)PF_EDGES * WROW, 0, 0);

    // D = A x B + C : one 16x16x4 f32 WMMA per edge per channel tile
    acc = __builtin_amdgcn_wmma_f32_16x16x4_f32(
        /*neg_a=*/false, A, /*neg_b=*/false, B,
        /*c_mod=*/(short)0, acc, /*reuse_a=*/false, /*reuse_b=*/false);
  }

  flush_tile(out, acc, cur, ctile, lane, (cur == r_first) || (cur == r_last));
}

extern "C" void kernel_launch(void* const* d_in, const int* in_sizes, int n_in,
                              void* d_out, int out_size, void* d_ws, size_t ws_size,
                              hipStream_t stream) {
  const float* node_feats = (const float*)d_in[0];
  const float* edge_attrs = (const float*)d_in[1];
  const float* tp_w       = (const float*)d_in[2];
  const int*   recv       = (const int*)d_in[3];
  float* out = (float*)d_out;

  const int nedges = in_sizes[1] / NLM;

  // Zero-init output (d_out is poisoned; zero-edge nodes must read 0,
  // boundary-node atomics accumulate onto 0).
  {
    long long n = (long long)out_size;
    int threads = 256;
    long long work = (n >> 2) + 1;
    int blocks = (int)((work + threads - 1) / threads);
    imptp_zero<<<blocks, threads, 0, stream>>>(out, n);
  }

  {
    int blocks = (nedges + EPB - 1) / EPB;
    imptp_main<<<blocks, 128, 0, stream>>>(node_feats, edge_attrs, tp_w, recv,
                                           out, nedges);
  }
}